// FastFoodTransform_29781303230782
// MI455X (gfx1250) — compile-verified
//
#include <hip/hip_runtime.h>

typedef __attribute__((ext_vector_type(2))) float v2f;
typedef __attribute__((ext_vector_type(8))) float v8f;

#define G 128            // block_mat is 128x128 (K = N = 128)
#define LDS_STRIDE 132   // padded k-stride for transposed B (132 % 64 == 4 -> conflict-free b64 reads)
#define ROWS_PER_WAVE 16
#define WAVES_PER_BLOCK 8
#define ROWS_PER_BLOCK (ROWS_PER_WAVE * WAVES_PER_BLOCK)  // 128

// Y[R,128] = X[R,128] * M[128,128], R = 4*4096*32.
// One wave computes a 16x128 output strip via V_WMMA_F32_16X16X4_F32.
__global__ __launch_bounds__(256) void blockdiag_wmma_f32(
    const float* __restrict__ x, const float* __restrict__ m, float* __restrict__ y)
{
    // B^T in LDS, padded: bt[n*132 + k] = M[k][n]. 128*132*4 = 67,584 B (< 320 KB WGP LDS).
    __shared__ float bt[G * LDS_STRIDE];

    const int tid = threadIdx.x;

    // Cooperative fill: coalesced b128 global reads of M, transposed scatter into LDS.
    for (int i = tid; i < (G * G) / 4; i += 256) {
        const int k  = i >> 5;         // 0..127  (row of M)
        const int n4 = (i & 31) * 4;   // 0,4,...,124 (col of M)
        const float4 v = *(const float4*)(m + k * G + n4);
        bt[(n4 + 0) * LDS_STRIDE + k] = v.x;
        bt[(n4 + 1) * LDS_STRIDE + k] = v.y;
        bt[(n4 + 2) * LDS_STRIDE + k] = v.z;
        bt[(n4 + 3) * LDS_STRIDE + k] = v.w;
    }
    __syncthreads();

    const int wave = tid >> 5;
    const int lane = tid & 31;
    const int lo   = lane & 15;   // row (A/C) or col (B) within tile
    const int hi   = lane >> 4;   // 0: lanes 0-15, 1: lanes 16-31

    const long rowBase = (long)blockIdx.x * ROWS_PER_BLOCK + (long)wave * ROWS_PER_WAVE;

    // A-fragment base: 16x4 f32 A layout — lanes 0-15 hold (K0,K1), lanes 16-31 hold (K2,K3)
    // of rows lane&15. Per k-step: one 8-byte load, 8B-aligned.
    const float* aRow = x + (rowBase + lo) * (long)G + hi * 2;

    v8f acc[8] = {};   // 8 N-tiles of 16x16 f32 accumulators

    #pragma unroll
    for (int kk = 0; kk < G / 4; ++kk) {
        const v2f a = *(const v2f*)(aRow + kk * 4);
        #pragma unroll
        for (int nt = 0; nt < 8; ++nt) {
            // B-fragment: 4x16 f32 B layout — lanes 0-15 hold rows (K0,K1), lanes 16-31 (K2,K3),
            // col = lane&15. Transposed LDS makes (k,k+1) contiguous -> single ds_load_b64.
            const v2f b = *(const v2f*)&bt[(nt * 16 + lo) * LDS_STRIDE + kk * 4 + hi * 2];
            acc[nt] = __builtin_amdgcn_wmma_f32_16x16x4_f32(
                /*neg_a=*/false, a, /*neg_b=*/false, b,
                /*c_mod=*/(short)0, acc[nt], /*reuse_a=*/false, /*reuse_b=*/false);
        }
    }

    // C/D layout: VGPR j = row (j + hi*8), col = lane&15. Contiguous 64B runs per half-wave.
    #pragma unroll
    for (int nt = 0; nt < 8; ++nt) {
        float* o = y + (rowBase + hi * 8) * (long)G + nt * 16 + lo;
        #pragma unroll
        for (int j = 0; j < 8; ++j)
            o[(long)j * G] = acc[nt][j];
    }
}

extern "C" void kernel_launch(void* const* d_in, const int* in_sizes, int n_in,
                              void* d_out, int out_size, void* d_ws, size_t ws_size,
                              hipStream_t stream) {
    const float* x = (const float*)d_in[0];   // (4, 4096, 4096) f32
    const float* m = (const float*)d_in[1];   // (128, 128) f32
    float* y = (float*)d_out;                 // (4, 4096, 4096) f32

    const long total = (long)in_sizes[0];         // 67,108,864
    const long rows  = total / G;                 // 524,288
    const int  blocks = (int)(rows / ROWS_PER_BLOCK);  // 4096 (exact for these shapes)

    blockdiag_wmma_f32<<<blocks, 256, 0, stream>>>(x, m, y);
}